// AcousticGuitarPoC_69947837383230
// MI455X (gfx1250) — compile-verified
//
#include <hip/hip_runtime.h>

typedef __attribute__((ext_vector_type(2))) float v2f;
typedef __attribute__((ext_vector_type(8))) float v8f;
typedef __attribute__((ext_vector_type(4))) unsigned int u32x4;
typedef __attribute__((ext_vector_type(8))) int i32x8;
typedef __attribute__((ext_vector_type(4))) int i32x4;

#define FS_F 44100.0f
#define PI_F 3.14159265358979f
#define N 32768
#define NB 24
#define NBLK (N / 16)    // 2048 16-sample blocks
#define NTILE (N / 256)  // 128 WMMA tiles (16 blocks each)
#define WAVES 8
#define BLOCK 256

// ---------------------------------------------------------------------------
// Kernel 1: per-batch parameter MLP + derived DSP coefficients (tiny).
// params[b*16 + {0..8}] = scale_pre, c_pre, scale_post, c_post, g, w1,w2,w3, Di
// Also writes the 3 scalar means (loop_coeff/mix/post) to out_scalars[0..2].
// ---------------------------------------------------------------------------
__global__ void ks_params(const float* __restrict__ pitch,
                          const float* __restrict__ w1, const float* __restrict__ b1,
                          const float* __restrict__ w2, const float* __restrict__ b2,
                          const float* __restrict__ gain,
                          float* __restrict__ params, float* __restrict__ out_scalars) {
  __shared__ float tmp[8][3];
  int b = threadIdx.x;
  if (b < 8) {
    float p = pitch[b];
    float h[16];
#pragma unroll
    for (int j = 0; j < 16; ++j) {
      float v = p * w1[j] + b1[j];
      h[j] = v > 0.f ? v : 0.f;
    }
    float m[3];
#pragma unroll
    for (int o = 0; o < 3; ++o) {
      float acc = b2[o];
#pragma unroll
      for (int j = 0; j < 16; ++j) acc += w2[o * 16 + j] * h[j];
      m[o] = acc;
    }
    float lc = fminf(fmaxf(m[0], -2.0f), 2.5f);
    float lm = fminf(fmaxf(m[1], -2.5f), 0.0f);
    float lp = fminf(fmaxf(m[2], -4.0f), 4.0f);
    float g = 0.999f / (1.f + __expf(-lc));
    float s = 1.f / (1.f + __expf(-lm));
    float f0 = fmaxf(p, 60.f);
    float D = fminf(fmaxf(FS_F / f0, 2.f), 735.f);
    float Di = floorf(D);
    float fr = D - Di;
    float mult = fminf(fmaxf(2.f + 6.f * (f0 - 60.f) / 600.f, 2.f), 8.f);
    float cutoff = fminf(2.f * PI_F * f0 * mult / FS_F, PI_F * 0.9f);
    float alpha = 1.f - __expf(-cutoff);
    float cutp = fminf(PI_F / (1.f + __expf(-lp)), PI_F * 0.99f);
    float ap = 1.f - __expf(-cutp);
    float* P = params + b * 16;
    P[0] = gain[0] * alpha;  // pre-scale applied to excitation
    P[1] = 1.f - alpha;      // one-pole feedback coeff (pre)
    P[2] = ap;               // post-scale
    P[3] = 1.f - ap;         // one-pole feedback coeff (post)
    P[4] = g;
    P[5] = (1.f - s) * (1.f - fr);            // KS tap weight y[n-Di]
    P[6] = (1.f - s) * fr + s * (1.f - fr);   // KS tap weight y[n-Di-1]
    P[7] = s * fr;                            // KS tap weight y[n-Di-2]
    P[8] = Di;
    tmp[b][0] = lc; tmp[b][1] = lm; tmp[b][2] = lp;
  }
  __syncthreads();
  if (threadIdx.x == 0) {
    for (int k = 0; k < 3; ++k) {
      float a = 0.f;
      for (int i = 0; i < 8; ++i) a += tmp[i][k];
      out_scalars[k] = a * 0.125f;
    }
  }
}

// ---------------------------------------------------------------------------
// Blocked one-pole IIR y[n] = x[n] + c*y[n-1], in place over xs[0..N).
// Phase 1: all 2048 block partials in parallel via WMMA (8 waves x 16 tiles),
//          A = lower-triangular Toeplitz T[i][j]=c^{i-j}, hoisted into VGPRs.
// Phase 2: 2048-step scalar carry scan with coefficient c^16.
// Phase 3: parallel fix-up  y[16b+i] = partial + c^{i+1}*carry[b]  (LDS table).
// ---------------------------------------------------------------------------
__device__ void onepole_blocked(float* __restrict__ xs, float* __restrict__ carry,
                                float* __restrict__ cpow, float c, int tid) {
  // cpow[t] = c^t for t in [0,16], built without dynamic private indexing
  if (tid < 17) {
    float p = 1.f;
#pragma unroll
    for (int j = 0; j < 16; ++j) p = (j < tid) ? p * c : p;
    cpow[tid] = p;
  }
  __syncthreads();

  int lane = tid & 31, wave = tid >> 5;
  int row = lane & 15, mofs = (lane < 16) ? 0 : 8;
  int kofs = (lane < 16) ? 0 : 2;

  // Hoisted, tile-invariant A panels (registers): T[row][k0], T[row][k0+1]
  v2f apan[4];
#pragma unroll
  for (int kk = 0; kk < 4; ++kk) {
    int k0 = 4 * kk + kofs;
    apan[kk].x = (row >= k0)     ? cpow[row - k0]     : 0.f;
    apan[kk].y = (row >= k0 + 1) ? cpow[row - k0 - 1] : 0.f;
  }
  float c16 = cpow[16];

  for (int tile = wave; tile < NTILE; tile += WAVES) {
    int base = tile * 256;
    v8f d = {};
#pragma unroll
    for (int kk = 0; kk < 4; ++kk) {
      int k0 = 4 * kk + kofs;
      v2f bv;
      bv.x = xs[base + row * 16 + k0];       // X[k][col], col = lane&15
      bv.y = xs[base + row * 16 + k0 + 1];
      d = __builtin_amdgcn_wmma_f32_16x16x4_f32(false, apan[kk], false, bv,
                                                (short)0, d, false, false);
    }
#pragma unroll
    for (int v = 0; v < 8; ++v) xs[base + row * 16 + v + mofs] = d[v];
  }
  __syncthreads();

  if (tid == 0) {
    float cy = 0.f;
    for (int b2 = 0; b2 < NBLK; ++b2) {
      carry[b2] = cy;                    // y entering block b2 (= y[16*b2-1])
      cy = xs[b2 * 16 + 15] + c16 * cy;
    }
  }
  __syncthreads();
  for (int n = tid; n < N; n += BLOCK) {
    int i = n & 15, b2 = n >> 4;
    xs[n] += cpow[i + 1] * carry[b2];    // c^{i+1}*carry, LDS-table lookup
  }
  __syncthreads();
}

// ---------------------------------------------------------------------------
// Karplus-Strong feedback loop, chunk-parallel: delay Di >= 66 samples means
// any chunk of <= Di consecutive samples depends only on pre-chunk history.
// ---------------------------------------------------------------------------
__device__ void ks_loop(float* __restrict__ xs, float g, float wk1, float wk2,
                        float wk3, int Di, int tid) {
  int chunk = Di < BLOCK ? Di : BLOCK;
  if (chunk < 1) chunk = 1;
  for (int t = 0; t < N; t += chunk) {
    int n = t + tid;
    if (tid < chunk && n < N) {
      int j = n - Di;
      float b1v = (j >= 0) ? xs[j] : 0.f;
      float b2v = (j - 1 >= 0) ? xs[j - 1] : 0.f;
      float b3v = (j - 2 >= 0) ? xs[j - 2] : 0.f;
      xs[n] = xs[n] + g * (wk1 * b1v + wk2 * b2v + wk3 * b3v);
    }
    __syncthreads();
  }
}

// ---------------------------------------------------------------------------
// Main per-batch synthesis kernel. Whole signal lives in LDS (128 KB of the
// 320 KB WGP pool); excitation staged in via one TDM tensor load.
// ---------------------------------------------------------------------------
__global__ void ks_synth(const float* __restrict__ exc,
                         const float* __restrict__ params,
                         const float* __restrict__ gains,
                         float* __restrict__ out) {
  extern __shared__ float smem[];
  float* xs = smem;                 // [32768] working signal
  float* carry = xs + N;            // [2048]  one-pole block carries
  float* wsum = carry + NBLK;       // [2048]  per-wave band sums (8 x 256)
  float* sL = wsum + 2048;          // [128]   per-wave s[15] per block
  float* sP = sL + 128;             // [128]   per-wave s[14] per block
  float* st1 = sP + 128;            // [128]   entering state y[-1] per block
  float* st2 = st1 + 128;           // [128]   entering state y[-2] per block
  float* bandCarry = st2 + 128;     // [48]    (y1,y2) per band across tiles
  float* cpow = bandCarry + 48;     // [17]    one-pole power table
  float* hTab = cpow + 17;          // [24*16] band impulse responses
  float* phiTab = hTab + NB * 16;   // [24*16] response to y[-1]=1
  float* psiTab = phiTab + NB * 16; // [24*16] response to y[-2]=1

  int b = blockIdx.x;
  int tid = threadIdx.x;
  int lane = tid & 31, wave = tid >> 5;
  int row = lane & 15, mofs = (lane < 16) ? 0 : 8;
  int kofs = (lane < 16) ? 0 : 2;

  const float* P = params + b * 16;
  float scale_pre = P[0], c_pre = P[1], scale_post = P[2], c_post = P[3];
  float g = P[4], wk1 = P[5], wk2 = P[6], wk3 = P[7];
  int Di = (int)P[8];

  // --- Stage A: TDM async DMA of excitation[b][:] into LDS (wave 0 issues) ---
  if (tid < 32) {
    unsigned long long gaddr =
        (unsigned long long)(const void*)(exc + (size_t)b * N);
    unsigned int lds_off = (unsigned int)(unsigned long long)(void*)xs;
    u32x4 g0;
    i32x8 g1;
    i32x4 g2 = {0, 0, 0, 0}, g3 = {0, 0, 0, 0};
    i32x8 g4 = {0, 0, 0, 0, 0, 0, 0, 0};
    g0[0] = 1u;                                        // count=1, no gather
    g0[1] = lds_off;                                   // lds_addr
    g0[2] = (unsigned int)(gaddr & 0xffffffffu);       // global_addr[31:0]
    g0[3] = (unsigned int)((gaddr >> 32) & 0x1ffffffu) // global_addr[56:32]
            | (2u << 30);                              // type = 2 (image)
    g1[0] = (int)(2u << 16);                           // data_size = 4 bytes
    g1[1] = (int)((unsigned)(N & 0xffff) << 16);       // tensor_dim0[15:0]
    g1[2] = (int)(((unsigned)N >> 16) & 0xffffu)       // tensor_dim0[31:16]
            | (1 << 16);                               // tensor_dim1 = 1
    g1[3] = (int)((unsigned)(N & 0xffff) << 16);       // tile_dim0 = N
    g1[4] = 1;                                         // tile_dim1=1, tile_dim2=0
    g1[5] = N;                                         // tensor_dim0_stride
    g1[6] = 0;
    g1[7] = 0;
    __builtin_amdgcn_tensor_load_to_lds(g0, g1, g2, g3, g4, 0);
    __builtin_amdgcn_s_wait_tensorcnt(0);
  }
  __syncthreads();

  // scale excitation: x = exc * gain * alpha
  for (int n = tid; n < N; n += BLOCK) xs[n] *= scale_pre;
  __syncthreads();

  // --- Stage B: pre-KS one-pole lowpass (WMMA-blocked scan) ---
  onepole_blocked(xs, carry, cpow, c_pre, tid);

  // --- Stage C: Karplus-Strong fractional-delay feedback loop ---
  ks_loop(xs, g, wk1, wk2, wk3, Di, tid);

  // --- Stage D: post one-pole lowpass (WMMA-blocked scan) ---
  for (int n = tid; n < N; n += BLOCK) xs[n] *= scale_post;
  __syncthreads();
  onepole_blocked(xs, carry, cpow, c_post, tid);

  // --- Stage E: 24-band two-pole resonator body ---
  // Precompute per-band tables into LDS (one thread per band, 16-step
  // recurrences in registers, sequential stores -- no dynamic private arrays).
  if (tid < NB) {
    int k = tid;
    float fc = 80.f * __expf((float)k * (__logf(100.f) / 23.f));
    float w = 2.f * PI_F * fc / FS_F;
    float r = __expf(-PI_F * fc / (10.f * FS_F));
    float a1 = -2.f * r * __cosf(w);
    float a2 = r * r;
    float b0 = 1.f - r;
    float hm1 = 0.f, hm2 = 0.f;
    float p1 = 1.f, p2 = 0.f;
    float q1 = 0.f, q2 = 1.f;
#pragma unroll
    for (int i = 0; i < 16; ++i) {
      float hv = (i == 0 ? b0 : 0.f) - a1 * hm1 - a2 * hm2;
      hTab[k * 16 + i] = hv; hm2 = hm1; hm1 = hv;
      float pv = -a1 * p1 - a2 * p2;
      phiTab[k * 16 + i] = pv; p2 = p1; p1 = pv;
      float qv = -a1 * q1 - a2 * q2;
      psiTab[k * 16 + i] = qv; q2 = q1; q1 = qv;
    }
  }
  if (tid < NB * 2) bandCarry[tid] = 0.f;
  __syncthreads();

  // Hoist all tile-invariant band data into registers: A panels (Toeplitz of
  // hTab), apply coefficients phi/psi at this lane's M rows, scan-edge values.
  v2f aq[3][4];
  float phv[3][8], psv[3][8];
  float ph15[3], ph14[3], ps15[3], ps14[3], gv[3];
#pragma unroll
  for (int q = 0; q < 3; ++q) {
    int k = wave * 3 + q;
    gv[q] = gains[k];
#pragma unroll
    for (int kk = 0; kk < 4; ++kk) {
      int k0 = 4 * kk + kofs;
      aq[q][kk].x = (row >= k0)     ? hTab[k * 16 + row - k0]     : 0.f;
      aq[q][kk].y = (row >= k0 + 1) ? hTab[k * 16 + row - k0 - 1] : 0.f;
    }
#pragma unroll
    for (int v = 0; v < 8; ++v) {
      phv[q][v] = phiTab[k * 16 + v + mofs];
      psv[q][v] = psiTab[k * 16 + v + mofs];
    }
    ph15[q] = phiTab[k * 16 + 15];
    ph14[q] = phiTab[k * 16 + 14];
    ps15[q] = psiTab[k * 16 + 15];
    ps14[q] = psiTab[k * 16 + 14];
  }
  __syncthreads();

  for (int tile = 0; tile < NTILE; ++tile) {
    int base = tile * 256;
    v8f acc = {};
#pragma unroll
    for (int q = 0; q < 3; ++q) {
      v8f d = {};
#pragma unroll
      for (int kk = 0; kk < 4; ++kk) {
        int k0 = 4 * kk + kofs;
        v2f bv;
        bv.x = xs[base + row * 16 + k0];
        bv.y = xs[base + row * 16 + k0 + 1];
        d = __builtin_amdgcn_wmma_f32_16x16x4_f32(false, aq[q][kk], false, bv,
                                                  (short)0, d, false, false);
      }
      if (lane >= 16) {                           // publish s[15], s[14]
        sL[wave * 16 + row] = d[7];
        sP[wave * 16 + row] = d[6];
      }
      __syncthreads();
      if (lane == 0) {                            // 16-block 2-state carry scan
        int k = wave * 3 + q;
        float c1 = bandCarry[2 * k], c2 = bandCarry[2 * k + 1];
        for (int blk = 0; blk < 16; ++blk) {
          st1[wave * 16 + blk] = c1;
          st2[wave * 16 + blk] = c2;
          float s15 = sL[wave * 16 + blk], s14 = sP[wave * 16 + blk];
          float n1 = s15 + ph15[q] * c1 + ps15[q] * c2;
          float n2 = s14 + ph14[q] * c1 + ps14[q] * c2;
          c1 = n1; c2 = n2;
        }
        bandCarry[2 * k] = c1;
        bandCarry[2 * k + 1] = c2;
      }
      __syncthreads();
      float cs1 = st1[wave * 16 + row], cs2 = st2[wave * 16 + row];
#pragma unroll
      for (int v = 0; v < 8; ++v) {
        float y = d[v] + phv[q][v] * cs1 + psv[q][v] * cs2;
        acc[v] += gv[q] * y;
      }
      __syncthreads();
    }
    // each wave deposits its 3-band sum in D layout, then reduce 8 waves
#pragma unroll
    for (int v = 0; v < 8; ++v)
      wsum[wave * 256 + row * 16 + v + mofs] = acc[v];
    __syncthreads();
    float o = 0.f;
#pragma unroll
    for (int w = 0; w < WAVES; ++w) o += wsum[w * 256 + tid];
    out[(size_t)b * N + base + tid] = o;
    __syncthreads();
  }
}

// ---------------------------------------------------------------------------
extern "C" void kernel_launch(void* const* d_in, const int* in_sizes, int n_in,
                              void* d_out, int out_size, void* d_ws,
                              size_t ws_size, hipStream_t stream) {
  const float* exc = (const float*)d_in[0];     // [8,1,32768]
  const float* pitch = (const float*)d_in[1];   // [8,1]
  const float* w1 = (const float*)d_in[2];      // [16,1]
  const float* b1 = (const float*)d_in[3];      // [16]
  const float* w2 = (const float*)d_in[4];      // [3,16]
  const float* b2 = (const float*)d_in[5];      // [3]
  const float* gain = (const float*)d_in[6];    // scalar
  const float* gains = (const float*)d_in[7];   // [1,24]
  float* out = (float*)d_out;                   // 8*32768 + 3 scalars
  float* params = (float*)d_ws;

  ks_params<<<1, 32, 0, stream>>>(pitch, w1, b1, w2, b2, gain, params,
                                  out + (size_t)8 * N);

  size_t shbytes =
      (size_t)(N + NBLK + 2048 + 4 * 128 + 48 + 17 + 3 * NB * 16) * sizeof(float);
  ks_synth<<<8, BLOCK, shbytes, stream>>>(exc, params, gains, out);
}